// GNN_90744069030651
// MI455X (gfx1250) — compile-verified
//
#include <hip/hip_runtime.h>
#include <math.h>

#define D 64

typedef __attribute__((ext_vector_type(2))) float v2f;
typedef __attribute__((ext_vector_type(8))) float v8f;
typedef __attribute__((ext_vector_type(4))) int   v4i;

// gfx1250 async global->LDS copy path (ASYNCcnt), if the toolchain exposes it.
#if defined(__HIP_DEVICE_COMPILE__) &&                                         \
    __has_builtin(__builtin_amdgcn_global_load_async_to_lds_b128) &&           \
    __has_builtin(__builtin_amdgcn_s_wait_asynccnt)
#define GAT_ASYNC_LDS 1
#else
#define GAT_ASYNC_LDS 0
#endif

// ---------------------------------------------------------------------------
// Float atomic max via monotonic int mapping (init with -inf).
// ---------------------------------------------------------------------------
__device__ __forceinline__ void atomicMaxF(float* addr, float v) {
  if (v >= 0.0f)
    atomicMax((int*)addr, __float_as_int(v));
  else
    atomicMin((unsigned int*)addr, (unsigned int)__float_as_int(v));
}

// ---------------------------------------------------------------------------
// h = (RELU_IN ? relu(X) : X) @ W     [n x D] = [n x D][D x D]
// One block = 128 threads = 4 waves; block handles 16 rows, wave w handles
// columns [16w, 16w+16). K-loop of 16 x V_WMMA_F32_16X16X4_F32 (exact fp32).
// W staged in LDS (16 KB) via GLOBAL_LOAD_ASYNC_TO_LDS_B128 when available.
// ---------------------------------------------------------------------------
template <bool RELU_IN>
__global__ void __launch_bounds__(128)
gat_gemm_wmma(const float* __restrict__ X, const float* __restrict__ W,
              float* __restrict__ H, int n) {
  __shared__ float Wl[D * D];

#if GAT_ASYNC_LDS
  {
    // 16 KB = 1024 x 16B chunks; 128 threads x 8 async b128 transfers.
#pragma unroll
    for (int j = 0; j < (D * D) / 4 / 128; ++j) {
      const int c = (int)threadIdx.x + j * 128;  // 16-byte chunk index
      __builtin_amdgcn_global_load_async_to_lds_b128(
          (v4i*)(W + c * 4), (v4i*)(Wl + c * 4),
          /*imm offset=*/0, /*cpol=*/0);
    }
    __builtin_amdgcn_s_wait_asynccnt(0);
  }
  __syncthreads();
#else
  for (int i = threadIdx.x; i < (D * D) / 4; i += blockDim.x)
    ((float4*)Wl)[i] = ((const float4*)W)[i];
  __syncthreads();
#endif

  const int lane  = threadIdx.x & 31;
  const int wave  = threadIdx.x >> 5;      // 0..3 -> column tile
  const int col0  = wave * 16;
  const int r     = lane & 15;             // row within A tile / col within B,C
  const int khalf = (lane >> 4) << 1;      // 0 or 2: K sub-pair per half-wave
  const int row0  = blockIdx.x * 16;

  int rowA = row0 + r;                     // clamp (no-op for n%16==0) keeps
  if (rowA >= n) rowA = n - 1;             // EXEC all-ones at the WMMA
  const float* xrow = X + (size_t)rowA * D;

  v8f acc = {};
#pragma unroll
  for (int k = 0; k < D; k += 4) {
    const int kk = k + khalf;
    float a0 = xrow[kk], a1 = xrow[kk + 1];
    if (RELU_IN) { a0 = fmaxf(a0, 0.0f); a1 = fmaxf(a1, 0.0f); }
    v2f a; a.x = a0; a.y = a1;
    v2f b; b.x = Wl[kk * D + col0 + r];
    /* */  b.y = Wl[(kk + 1) * D + col0 + r];
    acc = __builtin_amdgcn_wmma_f32_16x16x4_f32(
        /*neg_a=*/false, a, /*neg_b=*/false, b,
        /*c_mod=*/(short)0, acc, /*reuse_a=*/false, /*reuse_b=*/false);
  }

  // C/D layout: lanes 0-15 hold M = vgpr, lanes 16-31 hold M = vgpr + 8.
  const int rbase = row0 + ((lane >> 4) << 3);
#pragma unroll
  for (int v = 0; v < 8; ++v) {
    const int row = rbase + v;
    if (row < n) H[(size_t)row * D + col0 + r] = acc[v];
  }
}

// ---------------------------------------------------------------------------
// Per-node attention logits: asrc[i] = h[i].a_src, adst[i] = h[i].a_dst.
// One wave per node; also initializes the segment max (-inf) and sum (0).
// ---------------------------------------------------------------------------
__global__ void gat_node_logits(const float* __restrict__ h,
                                const float* __restrict__ a_src,
                                const float* __restrict__ a_dst,
                                float* __restrict__ asrc, float* __restrict__ adst,
                                float* __restrict__ m, float* __restrict__ s,
                                int n) {
  const int node = (int)((blockIdx.x * blockDim.x + threadIdx.x) >> 5);
  const int lane = threadIdx.x & 31;
  if (node >= n) return;
  const float h0 = h[(size_t)node * D + lane];
  const float h1 = h[(size_t)node * D + 32 + lane];
  float ps = h0 * a_src[lane] + h1 * a_src[32 + lane];
  float pd = h0 * a_dst[lane] + h1 * a_dst[32 + lane];
#pragma unroll
  for (int off = 16; off > 0; off >>= 1) {
    ps += __shfl_xor(ps, off);
    pd += __shfl_xor(pd, off);
  }
  if (lane == 0) {
    asrc[node] = ps;
    adst[node] = pd;
    m[node]    = -INFINITY;
    s[node]    = 0.0f;
  }
}

// ---------------------------------------------------------------------------
// Pass 1: e = leaky_relu(asrc[src] + adst[dst]); segment max into m[dst].
// Edges beyond nE are the implicit self-loops (i,i).
// ---------------------------------------------------------------------------
__global__ void gat_edge_max(const int* __restrict__ eSrc, const int* __restrict__ eDst,
                             const float* __restrict__ asrc, const float* __restrict__ adst,
                             float* __restrict__ ebuf, float* __restrict__ m,
                             int nE, int nNodes) {
  const int i = blockIdx.x * blockDim.x + threadIdx.x;
  const int tot = nE + nNodes;
  if (i >= tot) return;
  const int sIdx = (i < nE) ? eSrc[i] : (i - nE);
  const int dIdx = (i < nE) ? eDst[i] : (i - nE);
  float e = asrc[sIdx] + adst[dIdx];
  e = (e > 0.0f) ? e : 0.2f * e;          // leaky_relu, slope 0.2
  ebuf[i] = e;
  atomicMaxF(&m[dIdx], e);
}

// ---------------------------------------------------------------------------
// Pass 2: p = exp(e - m[dst]); segment sum into s[dst]; p cached in ebuf.
// ---------------------------------------------------------------------------
__global__ void gat_edge_expsum(const int* __restrict__ eDst,
                                float* __restrict__ ebuf,
                                const float* __restrict__ m, float* __restrict__ s,
                                int nE, int nNodes) {
  const int i = blockIdx.x * blockDim.x + threadIdx.x;
  const int tot = nE + nNodes;
  if (i >= tot) return;
  const int dIdx = (i < nE) ? eDst[i] : (i - nE);
  const float p = __expf(ebuf[i] - m[dIdx]);
  ebuf[i] = p;
  atomicAdd(&s[dIdx], p);
}

// ---------------------------------------------------------------------------
// out[i*D + k] = bias[k]   (bias pre-load so aggregation is pure atomic-add)
// ---------------------------------------------------------------------------
__global__ void gat_init_bias(float* __restrict__ out, const float* __restrict__ b,
                              int total) {
  const int i = blockIdx.x * blockDim.x + threadIdx.x;
  if (i < total) out[i] = b[i & (D - 1)];
}

// ---------------------------------------------------------------------------
// Pass 3: out[dst] += (p/s[dst]) * h[src].  One wave per edge; lane l handles
// features l and l+32 -> coalesced 256 B gather + global_atomic_add_f32.
// h and out both fit in L2 (12.8 MB each), so the random traffic stays on-die.
// ---------------------------------------------------------------------------
__global__ void gat_edge_aggregate(const int* __restrict__ eSrc, const int* __restrict__ eDst,
                                   const float* __restrict__ pbuf, const float* __restrict__ s,
                                   const float* __restrict__ h, float* __restrict__ out,
                                   int nE, int nNodes) {
  const int e    = (int)((blockIdx.x * blockDim.x + threadIdx.x) >> 5);
  const int lane = threadIdx.x & 31;
  const int tot  = nE + nNodes;
  if (e >= tot) return;
  const int sIdx = (e < nE) ? eSrc[e] : (e - nE);
  const int dIdx = (e < nE) ? eDst[e] : (e - nE);
  const float alpha = pbuf[e] / s[dIdx];
  const size_t hb = (size_t)sIdx * D;
  const size_t ob = (size_t)dIdx * D;
  atomicAdd(&out[ob + lane],      alpha * h[hb + lane]);
  atomicAdd(&out[ob + 32 + lane], alpha * h[hb + 32 + lane]);
}

// ---------------------------------------------------------------------------
// Orchestration
// ---------------------------------------------------------------------------
static void run_gat_layer(const float* x, const float* W, const float* av_src,
                          const float* av_dst, const float* bias, bool relu_in,
                          const int* eSrc, const int* eDst, int nE, int nNodes,
                          float* h, float* out, float* asrc, float* adst,
                          float* m, float* s, float* pbuf, hipStream_t stream) {
  const int rowTiles = (nNodes + 15) / 16;
  const int tot      = nE + nNodes;
  if (relu_in)
    gat_gemm_wmma<true><<<rowTiles, 128, 0, stream>>>(x, W, h, nNodes);
  else
    gat_gemm_wmma<false><<<rowTiles, 128, 0, stream>>>(x, W, h, nNodes);
  gat_node_logits<<<(nNodes + 7) / 8, 256, 0, stream>>>(h, av_src, av_dst,
                                                        asrc, adst, m, s, nNodes);
  gat_edge_max<<<(tot + 255) / 256, 256, 0, stream>>>(eSrc, eDst, asrc, adst,
                                                      pbuf, m, nE, nNodes);
  gat_edge_expsum<<<(tot + 255) / 256, 256, 0, stream>>>(eDst, pbuf, m, s, nE, nNodes);
  gat_init_bias<<<(nNodes * D + 255) / 256, 256, 0, stream>>>(out, bias, nNodes * D);
  gat_edge_aggregate<<<(tot + 7) / 8, 256, 0, stream>>>(eSrc, eDst, pbuf, s, h, out,
                                                        nE, nNodes);
}

extern "C" void kernel_launch(void* const* d_in, const int* in_sizes, int n_in,
                              void* d_out, int out_size, void* d_ws, size_t ws_size,
                              hipStream_t stream) {
  const float* embeds = (const float*)d_in[0];
  const int*   edges  = (const int*)d_in[1];      // [2, nE] int32
  const float* W1     = (const float*)d_in[2];
  const float* as1    = (const float*)d_in[3];
  const float* ad1    = (const float*)d_in[4];
  const float* b1     = (const float*)d_in[5];
  const float* W2     = (const float*)d_in[6];
  const float* as2    = (const float*)d_in[7];
  const float* ad2    = (const float*)d_in[8];
  const float* b2     = (const float*)d_in[9];

  const int nNodes = in_sizes[0] / D;             // 50000
  const int nE     = in_sizes[1] / 2;             // 800000
  const int* eSrc  = edges;
  const int* eDst  = edges + nE;

  // Workspace carve-up (floats): ~30 MB total
  float* ws   = (float*)d_ws;
  float* h    = ws;                               // [nNodes*D]
  float* o1   = h    + (size_t)nNodes * D;        // [nNodes*D]
  float* asrc = o1   + (size_t)nNodes * D;        // [nNodes]
  float* adst = asrc + nNodes;                    // [nNodes]
  float* m    = adst + nNodes;                    // [nNodes]
  float* s    = m    + nNodes;                    // [nNodes]
  float* pbuf = s    + nNodes;                    // [nE + nNodes]

  // Layer 1 (ReLU deferred into layer-2 GEMM A-load)
  run_gat_layer(embeds, W1, as1, ad1, b1, /*relu_in=*/false,
                eSrc, eDst, nE, nNodes, h, o1, asrc, adst, m, s, pbuf, stream);
  // Layer 2: x = relu(o1), result straight into d_out
  run_gat_layer(o1, W2, as2, ad2, b2, /*relu_in=*/true,
                eSrc, eDst, nE, nNodes, h, (float*)d_out, asrc, adst, m, s, pbuf,
                stream);
}